// CliffordSpectralConv3d_deprecated_73005854098022
// MI455X (gfx1250) — compile-verified
//
#include <hip/hip_runtime.h>
#include <hip/hip_bf16.h>
#include <math.h>

// ---------------------------------------------------------------------------
// Clifford spectral conv3d on gfx1250 (MI455X), all heavy math via
// V_WMMA_F32_16X16X4_F32.  B=4, C=32, S=64, M=12 (2M=24 corner modes/axis).
//
// Pipeline (all f32, complex kept as separate re/im planes):
//   per field f:  partial-DFT z -> y -> x   (GEMMs vs 24x64 twiddle tables)
//   core: per frequency point p: Out(256x4) = K(256x256,p) * In(256x4)
//   per output field j: inverse partial-DFT x -> y -> z (K=24 GEMMs),
//     replicating the upstream scatter bug as a kx permutation, 1/64^3
//     folded into the z-stage tables.
//
// Workspace requirement: ~252 MB (float): tables + ping-pong planes + corner
// spectra (8 planes) + core output.
// ---------------------------------------------------------------------------

typedef __attribute__((ext_vector_type(2))) float v2f;
typedef __attribute__((ext_vector_type(8))) float v8f;

static __device__ __forceinline__ v8f wmma4(v2f a, v2f b, v8f c) {
  // D(16x16,f32) = A(16x4,f32) * B(4x16,f32) + C
  return __builtin_amdgcn_wmma_f32_16x16x4_f32(false, a, false, b,
                                               (short)0, c, false, false);
}

static __device__ __forceinline__ int khat(int k) { return (k < 12) ? k : (k + 40); }

// Clifford blade product tables (Cl(3,0) multivector, upstream ordering).
__constant__ int cIDX[8][8] = {
  {0,1,2,3,4,5,6,7},{1,0,4,5,2,3,7,6},{2,4,0,6,1,7,3,5},{3,5,6,0,7,1,2,4},
  {4,2,1,7,0,6,5,3},{5,3,7,1,6,0,4,2},{6,7,3,2,5,4,0,1},{7,6,5,4,3,2,1,0}};
__constant__ float cSGN[8][8] = {
  { 1, 1, 1, 1,-1,-1,-1,-1},{ 1, 1,-1,-1, 1, 1,-1,-1},
  { 1, 1, 1,-1,-1, 1, 1, 1},{ 1, 1, 1, 1,-1,-1,-1,-1},
  { 1, 1,-1, 1, 1,-1, 1, 1},{ 1, 1,-1,-1, 1, 1,-1,-1},
  { 1, 1, 1,-1,-1, 1, 1, 1},{ 1, 1,-1, 1, 1,-1, 1, 1}};

// ---- workspace layout (float offsets) -------------------------------------
constexpr size_t OFF_TZ    = 0;            // [64][48]  fwd z twiddles (cos | -sin)
constexpr size_t OFF_TFRE  = 3072;         // [32][64]  fwd y/x: cos  (rows>=24 zero)
constexpr size_t OFF_TFIM  = 5120;         // [32][64]  -sin
constexpr size_t OFF_TFIMN = 7168;         // [32][64]  +sin
constexpr size_t OFF_TIRE  = 9216;         // [64][24]  inv x/y: cos
constexpr size_t OFF_TIIM  = 10752;        // [64][24]  +sin
constexpr size_t OFF_TIIMN = 12288;        // [64][24]  -sin
constexpr size_t OFF_TZI1  = 13824;        // [24][128] inv z: (cos | sin)  * 1/64^3
constexpr size_t OFF_TZI2  = 16896;        // [24][128] inv z: (-sin | cos) * 1/64^3
constexpr size_t OFF_BUFA  = 32768;        // 2 planes x 524288*24
constexpr size_t PA        = 12582912;
constexpr size_t OFF_BUFB  = OFF_BUFA + 2 * PA;   // 2 planes x 128*64*576
constexpr size_t PB        = 4718592;
constexpr size_t OFF_F     = OFF_BUFB + 2 * PB;   // 8 planes x 128*13824
constexpr size_t PF        = 1769472;
constexpr size_t OFF_G     = OFF_F + 8 * PF;      // 4*8*32*13824

// ---------------------------------------------------------------------------
__global__ void __launch_bounds__(256) k_init_tables(float* ws) {
  const float TP = 6.28318530717958647692f;
  const float inv64 = 1.0f / 64.0f;
  float* Tz = ws + OFF_TZ;
  for (int i = threadIdx.x; i < 64 * 48; i += blockDim.x) {
    int z = i / 48, k = i % 48;
    int kh = khat(k % 24);
    float ang = TP * (float)((z * kh) & 63) * inv64;
    Tz[i] = (k < 24) ? cosf(ang) : -sinf(ang);
  }
  float* Tfre = ws + OFF_TFRE; float* Tfim = ws + OFF_TFIM; float* Tfimn = ws + OFF_TFIMN;
  for (int i = threadIdx.x; i < 32 * 64; i += blockDim.x) {
    int m = i / 64, y = i % 64;
    float c = 0.f, s = 0.f;
    if (m < 24) {
      float ang = TP * (float)((y * khat(m)) & 63) * inv64;
      c = cosf(ang); s = sinf(ang);
    }
    Tfre[i] = c; Tfim[i] = -s; Tfimn[i] = s;
  }
  float* Tire = ws + OFF_TIRE; float* Tiim = ws + OFF_TIIM; float* Tiimn = ws + OFF_TIIMN;
  for (int i = threadIdx.x; i < 64 * 24; i += blockDim.x) {
    int x = i / 24, k = i % 24;
    float ang = TP * (float)((x * khat(k)) & 63) * inv64;
    Tire[i] = cosf(ang); Tiim[i] = sinf(ang); Tiimn[i] = -sinf(ang);
  }
  const float sc = 1.0f / 262144.0f;  // 1 / 64^3
  float* Tz1 = ws + OFF_TZI1; float* Tz2 = ws + OFF_TZI2;
  for (int i = threadIdx.x; i < 24 * 128; i += blockDim.x) {
    int k = i / 128, c = i % 128;
    int z = c & 63;
    float ang = TP * (float)((z * khat(k)) & 63) * inv64;
    float cs = cosf(ang) * sc, sn = sinf(ang) * sc;
    Tz1[i] = (c < 64) ? cs : sn;
    Tz2[i] = (c < 64) ? -sn : cs;
  }
}

// ---------------------------------------------------------------------------
// Forward stage Z: rows r=(b,c,x,y), contract z(64) -> 24 complex.
// out cols 0..23 = re (cos), 24..47 = im (-sin).
__global__ void __launch_bounds__(256) k_fwd_z(const float* __restrict__ x,
                                               const float* __restrict__ tz,
                                               float* __restrict__ o_re,
                                               float* __restrict__ o_im) {
  __shared__ float xs[128 * 65];
  __shared__ float ds[64 * 48];
  const int r0 = blockIdx.x * 128;
  for (int i = threadIdx.x; i < 128 * 64; i += 256) {
    int row = i >> 6, col = i & 63;
    xs[row * 65 + col] = x[(size_t)(r0 + row) * 64 + col];
  }
  for (int i = threadIdx.x; i < 64 * 48; i += 256) ds[i] = tz[i];
  __syncthreads();
  const int w = threadIdx.x >> 5, lane = threadIdx.x & 31;
  const int half = lane >> 4, ln = lane & 15;
  v8f acc0 = {}, acc1 = {}, acc2 = {};
  const int arow = (w * 16 + ln) * 65;
  for (int ks = 0; ks < 16; ++ks) {
    const int kk = ks * 4 + half * 2;
    v2f a; a.x = xs[arow + kk]; a.y = xs[arow + kk + 1];
    v2f b0, b1, b2;
    b0.x = ds[kk * 48 + ln];      b0.y = ds[(kk + 1) * 48 + ln];
    b1.x = ds[kk * 48 + 16 + ln]; b1.y = ds[(kk + 1) * 48 + 16 + ln];
    b2.x = ds[kk * 48 + 32 + ln]; b2.y = ds[(kk + 1) * 48 + 32 + ln];
    acc0 = wmma4(a, b0, acc0);
    acc1 = wmma4(a, b1, acc1);
    acc2 = wmma4(a, b2, acc2);
  }
  for (int v = 0; v < 8; ++v) {
    const size_t row = (size_t)(r0 + w * 16 + v + half * 8);
    o_re[row * 24 + ln] = acc0[v];
    const int c1 = 16 + ln;
    if (c1 < 24) o_re[row * 24 + c1] = acc1[v];
    else         o_im[row * 24 + (c1 - 24)] = acc1[v];
    o_im[row * 24 + 8 + ln] = acc2[v];
  }
}

// ---------------------------------------------------------------------------
// Forward stage Y: per slab (bc*64+x): out[ky(24),kz(24)] = D(24x64) x in[y,kz].
__global__ void __launch_bounds__(256) k_fwd_y(const float* __restrict__ i_re,
                                               const float* __restrict__ i_im,
                                               const float* __restrict__ t_re,
                                               const float* __restrict__ t_im,
                                               const float* __restrict__ t_imn,
                                               float* __restrict__ o_re,
                                               float* __restrict__ o_im) {
  __shared__ float sre[64 * 32], sim[64 * 32];
  __shared__ float tc[32 * 65], ts[32 * 65], tsn[32 * 65];
  const int slab = blockIdx.x;  // bc*64 + x
  for (int i = threadIdx.x; i < 64 * 32; i += 256) {
    int y = i >> 5, k = i & 31;
    float vr = 0.f, vi = 0.f;
    if (k < 24) {
      size_t off = ((size_t)slab * 64 + y) * 24 + k;
      vr = i_re[off]; vi = i_im[off];
    }
    sre[i] = vr; sim[i] = vi;
  }
  for (int i = threadIdx.x; i < 32 * 64; i += 256) {
    int m = i >> 6, yy = i & 63;
    tc[m * 65 + yy] = t_re[i]; ts[m * 65 + yy] = t_im[i]; tsn[m * 65 + yy] = t_imn[i];
  }
  __syncthreads();
  const int w = threadIdx.x >> 5, lane = threadIdx.x & 31;
  const int half = lane >> 4, ln = lane & 15;
  const int P = w >> 2, rt = (w >> 1) & 1, ct = w & 1;
  // P=0 (re): cos*re + sin*im ; P=1 (im): cos*im + (-sin)*re
  const float* A2 = P ? ts : tsn;
  const float* B1 = P ? sim : sre;
  const float* B2 = P ? sre : sim;
  v8f acc = {};
  const int arow = (rt * 16 + ln) * 65;
  for (int ks = 0; ks < 16; ++ks) {
    const int kk = ks * 4 + half * 2;
    v2f a1; a1.x = tc[arow + kk]; a1.y = tc[arow + kk + 1];
    v2f b1; b1.x = B1[kk * 32 + ct * 16 + ln]; b1.y = B1[(kk + 1) * 32 + ct * 16 + ln];
    acc = wmma4(a1, b1, acc);
    v2f a2; a2.x = A2[arow + kk]; a2.y = A2[arow + kk + 1];
    v2f b2; b2.x = B2[kk * 32 + ct * 16 + ln]; b2.y = B2[(kk + 1) * 32 + ct * 16 + ln];
    acc = wmma4(a2, b2, acc);
  }
  float* out = P ? o_im : o_re;
  const int kz = ct * 16 + ln;
  if (kz < 24) {
    for (int v = 0; v < 8; ++v) {
      const int ky = rt * 16 + v + half * 8;
      if (ky < 24) out[(size_t)slab * 576 + ky * 24 + kz] = acc[v];
    }
  }
}

// ---------------------------------------------------------------------------
// Forward stage X: per (bc, col-chunk of 64): out[kx(24), cols] = D x in[x, cols].
__global__ void __launch_bounds__(256) k_fwd_x(const float* __restrict__ i_re,
                                               const float* __restrict__ i_im,
                                               const float* __restrict__ t_re,
                                               const float* __restrict__ t_im,
                                               const float* __restrict__ t_imn,
                                               float* __restrict__ f_re,
                                               float* __restrict__ f_im) {
  __shared__ float sre[64 * 65], sim[64 * 65];
  __shared__ float tc[32 * 65], ts[32 * 65], tsn[32 * 65];
  const int bc = blockIdx.x / 9, chunk = blockIdx.x % 9;
  const int c0 = chunk * 64;
  for (int i = threadIdx.x; i < 64 * 64; i += 256) {
    int xr = i >> 6, j = i & 63;
    size_t off = ((size_t)bc * 64 + xr) * 576 + c0 + j;
    sre[xr * 65 + j] = i_re[off]; sim[xr * 65 + j] = i_im[off];
  }
  for (int i = threadIdx.x; i < 32 * 64; i += 256) {
    int m = i >> 6, yy = i & 63;
    tc[m * 65 + yy] = t_re[i]; ts[m * 65 + yy] = t_im[i]; tsn[m * 65 + yy] = t_imn[i];
  }
  __syncthreads();
  const int w = threadIdx.x >> 5, lane = threadIdx.x & 31;
  const int half = lane >> 4, ln = lane & 15;
  for (int it = 0; it < 2; ++it) {
    const int t = w * 2 + it;
    const int P = t >> 3, rt = (t >> 2) & 1, ct = t & 3;
    const float* A2 = P ? ts : tsn;
    const float* B1 = P ? sim : sre;
    const float* B2 = P ? sre : sim;
    v8f acc = {};
    const int arow = (rt * 16 + ln) * 65;
    for (int ks = 0; ks < 16; ++ks) {
      const int kk = ks * 4 + half * 2;
      v2f a1; a1.x = tc[arow + kk]; a1.y = tc[arow + kk + 1];
      v2f b1; b1.x = B1[kk * 65 + ct * 16 + ln]; b1.y = B1[(kk + 1) * 65 + ct * 16 + ln];
      acc = wmma4(a1, b1, acc);
      v2f a2; a2.x = A2[arow + kk]; a2.y = A2[arow + kk + 1];
      v2f b2; b2.x = B2[kk * 65 + ct * 16 + ln]; b2.y = B2[(kk + 1) * 65 + ct * 16 + ln];
      acc = wmma4(a2, b2, acc);
    }
    float* out = P ? f_im : f_re;
    const int col = c0 + ct * 16 + ln;
    for (int v = 0; v < 8; ++v) {
      const int kx = rt * 16 + v + half * 8;
      if (kx < 24) out[(size_t)bc * 13824 + (size_t)kx * 576 + col] = acc[v];
    }
  }
}

// ---------------------------------------------------------------------------
// Core blade-mixing einsum.  Per point p:  Out[(o,d),b] = sum_{i,c}
//   SGN[o,i]*W[IDX[o,i],c,d,p] * In[b,i,c,p].  32 consecutive points per block
//   so stride-13824 weight gathers reuse full cachelines.
__global__ void __launch_bounds__(256) k_core(const float* __restrict__ F,
                                              const float* __restrict__ W,
                                              float* __restrict__ G) {
  __shared__ float sB[256 * 4];
  const int p0 = blockIdx.x * 32;
  const int w = threadIdx.x >> 5, lane = threadIdx.x & 31;
  const int half = lane >> 4, ln = lane & 15;
  for (int pp = 0; pp < 32; ++pp) {
    const int p = p0 + pp;
    {  // load In vector for this point: k=(i,c), n=b
      const int k = threadIdx.x;
      const int i = k >> 5, c = k & 31;
      const float* pl = F + (size_t)i * PF;
      for (int b = 0; b < 4; ++b)
        sB[k * 4 + b] = pl[(size_t)(b * 32 + c) * 13824 + p];
    }
    __syncthreads();
    for (int rtl = 0; rtl < 2; ++rtl) {
      const int rt = w * 2 + rtl;
      const int grow = rt * 16 + ln;      // A-operand row for this lane
      const int o = grow >> 5, d = grow & 31;
      v8f acc = {};
      for (int ks = 0; ks < 64; ++ks) {
        const int kk = ks * 4 + half * 2;      // 0..254, even; never splits i
        const int i = kk >> 5;
        const int c = kk & 31;
        const int wi = cIDX[o][i];
        const float sg = cSGN[o][i];
        const size_t base = ((size_t)(wi * 32 + c) * 32 + d) * 13824 + p;
        if ((ks & 7) == 0 && ks + 8 < 64)
          __builtin_prefetch(W + base + (size_t)8 * 4 * 13824, 0, 0);
        v2f a; a.x = sg * W[base]; a.y = sg * W[base + (size_t)32 * 13824];
        v2f b;
        b.x = (ln < 4) ? sB[kk * 4 + ln] : 0.f;
        b.y = (ln < 4) ? sB[(kk + 1) * 4 + ln] : 0.f;
        acc = wmma4(a, b, acc);
      }
      if (ln < 4) {
        for (int v = 0; v < 8; ++v) {
          const int gr = rt * 16 + v + half * 8;
          const int oo = gr >> 5, dd = gr & 31;
          G[((size_t)(ln * 8 + oo) * 32 + dd) * 13824 + p] = acc[v];
        }
      }
    }
    __syncthreads();
  }
}

// ---------------------------------------------------------------------------
// Inverse stage X: per (bd, col-chunk 64): out[x(64), cols] = E(64x24) x H[kx, cols],
// with the upstream scatter bug replicated as a kx permutation on load.
__global__ void __launch_bounds__(256) k_inv_x(const float* __restrict__ G,
                                               int oRe, int oIm,
                                               const float* __restrict__ t_re,
                                               const float* __restrict__ t_im,
                                               const float* __restrict__ t_imn,
                                               float* __restrict__ o_re,
                                               float* __restrict__ o_im) {
  __shared__ float hre[24 * 64], him[24 * 64];
  __shared__ float tc[64 * 25], ts[64 * 25], tsn[64 * 25];
  const int bd = blockIdx.x / 9, chunk = blockIdx.x % 9;
  const int b = bd >> 5, d = bd & 31;
  const int c0 = chunk * 64;
  const size_t gre = ((size_t)(b * 8 + oRe) * 32 + d) * 13824;
  const size_t gim = ((size_t)(b * 8 + oIm) * 32 + d) * 13824;
  for (int i = threadIdx.x; i < 24 * 64; i += 256) {
    const int ka = i / 64, j = i % 64;
    const int col = c0 + j;
    const int ky = col / 24, kz = col % 24;
    // upstream bug: high-x / low-y / high-z octant sources from low-x
    const int kap = (ka >= 12 && ky < 12 && kz >= 12) ? (ka - 12) : ka;
    hre[ka * 64 + j] = G[gre + (size_t)kap * 576 + col];
    him[ka * 64 + j] = G[gim + (size_t)kap * 576 + col];
  }
  for (int i = threadIdx.x; i < 64 * 24; i += 256) {
    const int m = i / 24, k = i % 24;
    tc[m * 25 + k] = t_re[i]; ts[m * 25 + k] = t_im[i]; tsn[m * 25 + k] = t_imn[i];
  }
  __syncthreads();
  const int w = threadIdx.x >> 5, lane = threadIdx.x & 31;
  const int half = lane >> 4, ln = lane & 15;
  for (int it = 0; it < 4; ++it) {
    const int t = w * 4 + it;
    const int P = t >> 4, rt = (t >> 2) & 3, ct = t & 3;
    // P=0 (re): cos*re + (-sin)*im ; P=1 (im): cos*im + sin*re
    const float* A2 = P ? ts : tsn;
    const float* B1 = P ? him : hre;
    const float* B2 = P ? hre : him;
    v8f acc = {};
    const int arow = (rt * 16 + ln) * 25;
    for (int ks = 0; ks < 6; ++ks) {
      const int kk = ks * 4 + half * 2;
      v2f a1; a1.x = tc[arow + kk]; a1.y = tc[arow + kk + 1];
      v2f b1; b1.x = B1[kk * 64 + ct * 16 + ln]; b1.y = B1[(kk + 1) * 64 + ct * 16 + ln];
      acc = wmma4(a1, b1, acc);
      v2f a2; a2.x = A2[arow + kk]; a2.y = A2[arow + kk + 1];
      v2f b2; b2.x = B2[kk * 64 + ct * 16 + ln]; b2.y = B2[(kk + 1) * 64 + ct * 16 + ln];
      acc = wmma4(a2, b2, acc);
    }
    float* out = P ? o_im : o_re;
    const int col = c0 + ct * 16 + ln;
    for (int v = 0; v < 8; ++v) {
      const int xr = rt * 16 + v + half * 8;
      out[((size_t)bd * 64 + xr) * 576 + col] = acc[v];
    }
  }
}

// ---------------------------------------------------------------------------
// Inverse stage Y: per slab (bd*64+x): out[y(64), kz(24)] = E(64x24) x in[ky,kz].
__global__ void __launch_bounds__(256) k_inv_y(const float* __restrict__ i_re,
                                               const float* __restrict__ i_im,
                                               const float* __restrict__ t_re,
                                               const float* __restrict__ t_im,
                                               const float* __restrict__ t_imn,
                                               float* __restrict__ o_re,
                                               float* __restrict__ o_im) {
  __shared__ float sre[24 * 32], sim[24 * 32];
  __shared__ float tc[64 * 25], ts[64 * 25], tsn[64 * 25];
  const int slab = blockIdx.x;  // bd*64 + x
  for (int i = threadIdx.x; i < 24 * 32; i += 256) {
    const int ky = i >> 5, kz = i & 31;
    float vr = 0.f, vi = 0.f;
    if (kz < 24) {
      size_t off = (size_t)slab * 576 + ky * 24 + kz;
      vr = i_re[off]; vi = i_im[off];
    }
    sre[i] = vr; sim[i] = vi;
  }
  for (int i = threadIdx.x; i < 64 * 24; i += 256) {
    const int m = i / 24, k = i % 24;
    tc[m * 25 + k] = t_re[i]; ts[m * 25 + k] = t_im[i]; tsn[m * 25 + k] = t_imn[i];
  }
  __syncthreads();
  const int w = threadIdx.x >> 5, lane = threadIdx.x & 31;
  const int half = lane >> 4, ln = lane & 15;
  for (int it = 0; it < 2; ++it) {
    const int t = w * 2 + it;
    const int P = t >> 3, rt = (t >> 1) & 3, ct = t & 1;
    const float* A2 = P ? ts : tsn;
    const float* B1 = P ? sim : sre;
    const float* B2 = P ? sre : sim;
    v8f acc = {};
    const int arow = (rt * 16 + ln) * 25;
    for (int ks = 0; ks < 6; ++ks) {
      const int kk = ks * 4 + half * 2;
      v2f a1; a1.x = tc[arow + kk]; a1.y = tc[arow + kk + 1];
      v2f b1; b1.x = B1[kk * 32 + ct * 16 + ln]; b1.y = B1[(kk + 1) * 32 + ct * 16 + ln];
      acc = wmma4(a1, b1, acc);
      v2f a2; a2.x = A2[arow + kk]; a2.y = A2[arow + kk + 1];
      v2f b2; b2.x = B2[kk * 32 + ct * 16 + ln]; b2.y = B2[(kk + 1) * 32 + ct * 16 + ln];
      acc = wmma4(a2, b2, acc);
    }
    float* out = P ? o_im : o_re;
    const int kz = ct * 16 + ln;
    if (kz < 24) {
      for (int v = 0; v < 8; ++v) {
        const int y = rt * 16 + v + half * 8;
        out[((size_t)slab * 64 + y) * 24 + kz] = acc[v];
      }
    }
  }
}

// ---------------------------------------------------------------------------
// Inverse stage Z: rows r=(bd,x,y): out[r, z(64) complex] = in[r, kz(24)] x T(24x128),
// writes interleaved complex64 directly to d_out (scale 1/64^3 already in T).
__global__ void __launch_bounds__(256) k_inv_z(const float* __restrict__ i_re,
                                               const float* __restrict__ i_im,
                                               const float* __restrict__ tz1,
                                               const float* __restrict__ tz2,
                                               float* __restrict__ outf) {
  __shared__ float sre[128 * 25], sim[128 * 25];
  __shared__ float t1[24 * 128], t2[24 * 128];
  const int r0 = blockIdx.x * 128;
  for (int i = threadIdx.x; i < 128 * 24; i += 256) {
    const int row = i / 24, k = i % 24;
    sre[row * 25 + k] = i_re[(size_t)(r0 + row) * 24 + k];
    sim[row * 25 + k] = i_im[(size_t)(r0 + row) * 24 + k];
  }
  for (int i = threadIdx.x; i < 24 * 128; i += 256) { t1[i] = tz1[i]; t2[i] = tz2[i]; }
  __syncthreads();
  const int w = threadIdx.x >> 5, lane = threadIdx.x & 31;
  const int half = lane >> 4, ln = lane & 15;
  const int rt = w;  // 8 row tiles of 16
  v8f acc[8] = {};
  const int arow = (rt * 16 + ln) * 25;
  for (int ks = 0; ks < 6; ++ks) {
    const int kk = ks * 4 + half * 2;
    v2f a1; a1.x = sre[arow + kk]; a1.y = sre[arow + kk + 1];
    v2f a2; a2.x = sim[arow + kk]; a2.y = sim[arow + kk + 1];
    for (int ct = 0; ct < 8; ++ct) {
      v2f b1; b1.x = t1[kk * 128 + ct * 16 + ln]; b1.y = t1[(kk + 1) * 128 + ct * 16 + ln];
      acc[ct] = wmma4(a1, b1, acc[ct]);
      v2f b2; b2.x = t2[kk * 128 + ct * 16 + ln]; b2.y = t2[(kk + 1) * 128 + ct * 16 + ln];
      acc[ct] = wmma4(a2, b2, acc[ct]);
    }
  }
  for (int ct = 0; ct < 8; ++ct) {
    const int col = ct * 16 + ln;
    const int z = col & 63, isim = col >> 6;
    for (int v = 0; v < 8; ++v) {
      const size_t row = (size_t)(r0 + rt * 16 + v + half * 8);
      outf[(row * 64 + z) * 2 + isim] = acc[ct][v];
    }
  }
}

// ---------------------------------------------------------------------------
extern "C" void kernel_launch(void* const* d_in, const int* in_sizes, int n_in,
                              void* d_out, int out_size, void* d_ws, size_t ws_size,
                              hipStream_t stream) {
  (void)in_sizes; (void)n_in; (void)out_size; (void)ws_size;
  const float* X[4] = {(const float*)d_in[0], (const float*)d_in[1],
                       (const float*)d_in[2], (const float*)d_in[3]};
  const float* W = (const float*)d_in[4];   // [8,32,32,24,24,24]
  float* out = (float*)d_out;               // 4 x [B,C,64,64,64] complex64 (interleaved)
  float* ws = (float*)d_ws;

  float* Tz    = ws + OFF_TZ;
  float* Tfre  = ws + OFF_TFRE;  float* Tfim  = ws + OFF_TFIM;  float* Tfimn = ws + OFF_TFIMN;
  float* Tire  = ws + OFF_TIRE;  float* Tiim  = ws + OFF_TIIM;  float* Tiimn = ws + OFF_TIIMN;
  float* Tz1   = ws + OFF_TZI1;  float* Tz2   = ws + OFF_TZI2;
  float* bufA_re = ws + OFF_BUFA;          float* bufA_im = bufA_re + PA;
  float* bufB_re = ws + OFF_BUFB;          float* bufB_im = bufB_re + PB;
  float* F = ws + OFF_F;
  float* G = ws + OFF_G;

  k_init_tables<<<1, 256, 0, stream>>>(ws);

  for (int f = 0; f < 4; ++f) {
    k_fwd_z<<<4096, 256, 0, stream>>>(X[f], Tz, bufA_re, bufA_im);
    k_fwd_y<<<8192, 256, 0, stream>>>(bufA_re, bufA_im, Tfre, Tfim, Tfimn,
                                      bufB_re, bufB_im);
    // corner-spectrum plane order matches the multivector stacking:
    // i = f -> re plane, i = 7-f -> im plane.
    k_fwd_x<<<1152, 256, 0, stream>>>(bufB_re, bufB_im, Tfre, Tfim, Tfimn,
                                      F + (size_t)f * PF, F + (size_t)(7 - f) * PF);
  }

  k_core<<<432, 256, 0, stream>>>(F, W, G);

  const int ore[4] = {0, 1, 2, 3};
  const int oim[4] = {7, 6, 5, 4};
  for (int j = 0; j < 4; ++j) {
    k_inv_x<<<1152, 256, 0, stream>>>(G, ore[j], oim[j], Tire, Tiim, Tiimn,
                                      bufB_re, bufB_im);
    k_inv_y<<<8192, 256, 0, stream>>>(bufB_re, bufB_im, Tire, Tiim, Tiimn,
                                      bufA_re, bufA_im);
    k_inv_z<<<4096, 256, 0, stream>>>(bufA_re, bufA_im, Tz1, Tz2,
                                      out + (size_t)j * 67108864);
  }
}